// Contour_to_mask_4853313044679
// MI455X (gfx1250) — compile-verified
//
#include <hip/hip_runtime.h>
#include <hip/hip_bf16.h>

#define NPTS   128
#define IMSIZE 256
#define KSTEEP 100000.0f
#define EPSC   1e-5f
#define PI_F   3.14159265358979323846f

typedef __attribute__((ext_vector_type(2))) float v2f;
typedef __attribute__((ext_vector_type(8))) float v8f;

// Branch-free tanh. gfx1250 has a native V_TANH_F32 transcendental.
__device__ __forceinline__ float fast_tanh(float x) {
#if __has_builtin(__builtin_amdgcn_tanhf)
    return __builtin_amdgcn_tanhf(x);
#else
    // tanh(x) = sign(x) * (1 - t) / (1 + t),  t = exp(-2|x|); branch-free.
    const float a = __builtin_fabsf(x);
    const float t = __expf(-2.0f * a);
    const float r = (1.0f - t) * __builtin_amdgcn_rcpf(1.0f + t);
    return __builtin_copysignf(r, x);
#endif
}

// Branch-free acos, |x| <= 1 (guaranteed by the EPS clamp).
// Abramowitz & Stegun 4.4.46 degree-7: acos(a) = sqrt(1-a) * P(a), abs err ~2e-7.
__device__ __forceinline__ float fast_acos(float x) {
    const float a = __builtin_fabsf(x);
    float p = -0.0012624911f;
    p = __builtin_fmaf(p, a,  0.0066700901f);
    p = __builtin_fmaf(p, a, -0.0170881256f);
    p = __builtin_fmaf(p, a,  0.0308918810f);
    p = __builtin_fmaf(p, a, -0.0501743046f);
    p = __builtin_fmaf(p, a,  0.0889789874f);
    p = __builtin_fmaf(p, a, -0.2145988016f);
    p = __builtin_fmaf(p, a,  1.5707963050f);
    const float r = __builtin_amdgcn_sqrtf(1.0f - a) * p;   // acos(|x|)
    return (x >= 0.0f) ? r : (PI_F - r);                    // select, no branch
}

// One wave = 16 pixels. Per segment-group (4 segments), one
// V_WMMA_F32_16X16X4_F32 computes all {cross, dot, |diff|^2, |roll|^2}
// for 16 pixels x 4 segments; the tanh/acos epilogue is lane-local and
// branch-free, so EXEC stays all-ones through every WMMA.
__global__ __launch_bounds__(256) void
contour_to_mask_wmma(const float* __restrict__ contour,
                     float* __restrict__ out) {
    __shared__ float sc[NPTS * 2];   // this batch's contour (x,y interleaved)

    const int tid  = threadIdx.x;
    const int lane = tid & 31;
    const int wave = tid >> 5;

    const int gpix0 = blockIdx.x * 128;     // first global pixel of this block
    const int b     = gpix0 >> 16;          // 65536 pixels per batch image

    // stage contour[b] into LDS (256 floats, one per thread)
    sc[tid] = contour[b * (NPTS * 2) + tid];
    __syncthreads();

    const int halfhi = lane >> 4;           // 0: lanes 0-15, 1: lanes 16-31
    const int pixl   = lane & 15;           // pixel column N of this lane
    const int g      = gpix0 + wave * 16 + pixl;   // global output index
    const int pi     = g & 0xFFFF;                 // pixel within batch image
    const float x = (float)(pi >> 8)  * (1.0f / IMSIZE);   // mesh i / SIZE
    const float y = (float)(pi & 0xFF) * (1.0f / IMSIZE);  // mesh j / SIZE

    // B fragment: 4x16 pixel-feature matrix [1, x, y, x^2+y^2] (column = pixel).
    // Low-half lanes carry K=0 (v0) and K=1 (v1); high half K=2 / K=3.
    v2f bfrag;
    bfrag.x = halfhi ? y               : 1.0f;
    bfrag.y = halfhi ? (x * x + y * y) : x;

    // A-matrix row this lane supplies: M = seg_in_group*4 + quantity
    const int segl = pixl >> 2;   // 0..3: segment within group
    const int qty  = lane & 3;    // 0:cross 1:dot 2:|diff|^2 3:|roll|^2

    float acc = 0.0f;

    for (int grp = 0; grp < NPTS / 4; ++grp) {
        const int seg = grp * 4 + segl;
        const float px = sc[2 * seg];
        const float py = sc[2 * seg + 1];
        const int  nxi = (seg + 1) & (NPTS - 1);       // roll(-1)
        const float qx = sc[2 * nxi];
        const float qy = sc[2 * nxi + 1];

        // Coefficients of the quantity w.r.t. features [1, x, y, x^2+y^2]
        float k0, k1, k2, k3;
        if (qty == 0)      { k0 = py*qx - px*qy; k1 = qy - py;   k2 = px - qx;   k3 = 0.0f; }
        else if (qty == 1) { k0 = px*qx + py*qy; k1 = -(px+qx);  k2 = -(py+qy);  k3 = 1.0f; }
        else if (qty == 2) { k0 = px*px + py*py; k1 = -2.0f*px;  k2 = -2.0f*py;  k3 = 1.0f; }
        else               { k0 = qx*qx + qy*qy; k1 = -2.0f*qx;  k2 = -2.0f*qy;  k3 = 1.0f; }

        v2f afrag;                          // ISA 16x4 f32 A layout
        afrag.x = halfhi ? k2 : k0;         // K=2 : K=0
        afrag.y = halfhi ? k3 : k1;         // K=3 : K=1

        v8f c = {};
        // D = A x B   (8 args: neg_a, A, neg_b, B, c_mod, C, reuse_a, reuse_b)
        v8f d = __builtin_amdgcn_wmma_f32_16x16x4_f32(
            false, afrag, false, bfrag, (short)0, c, false, false);

        // Lane-local epilogue: this lane's 8 D values = 2 complete segments
        // (rows M = 8*halfhi + 0..7) for its pixel column.
        #pragma unroll
        for (int h = 0; h < 2; ++h) {
            const float cr = d[4 * h + 0];
            const float dt = d[4 * h + 1];
            const float n1 = d[4 * h + 2];
            const float n2 = d[4 * h + 3];
            const float sgn  = fast_tanh(KSTEEP * cr);
            const float prod = __builtin_fmaxf(n1 * n2, 1e-24f);
            float ca = dt * __builtin_amdgcn_rsqf(prod);
            ca = __builtin_fminf(__builtin_fmaxf(ca, -1.0f + EPSC), 1.0f - EPSC);
            acc = __builtin_fmaf(sgn, fast_acos(ca), acc);
        }
    }

    // Lane p and lane p+16 each summed 64 of the 128 segments for pixel p.
    acc += __shfl_xor(acc, 16, 32);

    if (lane < 16) {
        const float w = __builtin_fminf(
            __builtin_fabsf(acc) * 0.15915494309189535f, 1.0f);   // /(2*pi), clip
        out[g] = w;
    }
}

extern "C" void kernel_launch(void* const* d_in, const int* in_sizes, int n_in,
                              void* d_out, int out_size, void* d_ws, size_t ws_size,
                              hipStream_t stream) {
    (void)n_in; (void)d_ws; (void)ws_size; (void)out_size;
    const float* contour = (const float*)d_in[0];
    float* out = (float*)d_out;

    const int nbatch  = in_sizes[0] / (NPTS * 2);          // B = 2 in reference
    const int npixels = nbatch * IMSIZE * IMSIZE;          // pixels across batches
    const int nblocks = npixels / 128;                     // 128 pixels per 256-thr block

    contour_to_mask_wmma<<<nblocks, 256, 0, stream>>>(contour, out);
}